// BotRGCN_27264452395299
// MI455X (gfx1250) — compile-verified
//
#include <hip/hip_runtime.h>
#include <math.h>

#define NN 50000
#define EE 600000
#define DD 128
#define QQ 32
#define KDESC 768
#define EPSV 1e-5f
#define RBLK 512

typedef __attribute__((ext_vector_type(2))) float v2f;
typedef __attribute__((ext_vector_type(8))) float v8f;
typedef int v4i __attribute__((vector_size(16)));

typedef __attribute__((address_space(1))) v4i* gas_v4i;
typedef __attribute__((address_space(3))) v4i* las_v4i;

__device__ __forceinline__ float leakyf(float x) { return x > 0.f ? x : 0.01f * x; }

// Async DMA: 16B global -> LDS, tracked by ASYNCcnt (CDNA5 GLOBAL_LOAD_ASYNC_TO_LDS_B128).
__device__ __forceinline__ void async_copy16(const void* g, const void* l) {
    __builtin_amdgcn_global_load_async_to_lds_b128(
        (gas_v4i)(uintptr_t)g, (las_v4i)(unsigned)(uintptr_t)l, 0, 0);
}
__device__ __forceinline__ void wait_async0() {
    asm volatile("s_wait_asynccnt 0x0" ::: "memory");
}

__device__ __forceinline__ v8f wmma4(v2f a, v2f b, v8f c) {
    return __builtin_amdgcn_wmma_f32_16x16x4_f32(false, a, false, b, (short)0, c, false, false);
}

// Monotone float -> uint mapping so unsigned atomic max == float max (handles negatives).
__device__ __forceinline__ unsigned mapf(float f) {
    unsigned b = __float_as_uint(f);
    return (b & 0x80000000u) ? ~b : (b | 0x80000000u);
}
__device__ __forceinline__ float unmapf(unsigned u) {
    unsigned b = (u & 0x80000000u) ? (u ^ 0x80000000u) : ~u;
    return __uint_as_float(b);
}
#define AGG_INIT 0x007FFFFFu   // mapf(-inf)

// ---------------------------------------------------------------------------
// Fused feature projection: x[N,128] = leaky([desc@Wd+bd | tw@Wt+bt | num@Wn+bn | cat@Wc+bc])
// Block: 128 thr = 4 waves, 32 rows; wave = (rowHalf, matrix) -> desc/tweet touched ONCE from HBM.
// ---------------------------------------------------------------------------
__global__ __launch_bounds__(128)
void proj_kernel(const float* __restrict__ desc, const float* __restrict__ tw,
                 const float* __restrict__ numf, const float* __restrict__ catf,
                 const float* __restrict__ Wd, const float* __restrict__ bd,
                 const float* __restrict__ Wt, const float* __restrict__ bt,
                 const float* __restrict__ Wn, const float* __restrict__ bn,
                 const float* __restrict__ Wc, const float* __restrict__ bc,
                 float* __restrict__ x)
{
    const int tid  = threadIdx.x;
    const int lane = tid & 31;
    const int wid  = tid >> 5;            // 0..3
    const int rowBlk = blockIdx.x * 32;
    const int half = wid >> 1;            // row half within block
    const int mat  = wid & 1;             // 0 = desc, 1 = tweet
    const int row0 = rowBlk + half * 16;

    if (row0 < NN) {                       // N % 16 == 0 -> full tile or none
        const float* A    = mat ? tw : desc;
        const float* W    = mat ? Wt : Wd;
        const float* bias = mat ? bt : bd;
        const int colBase = mat * 32;
        const int n  = lane & 15;          // N index / A-row index (M)
        const int hi = lane >> 4;          // lane half selects K pair

        v8f c0, c1;
        float bv0 = bias[n], bv1 = bias[16 + n];
#pragma unroll
        for (int v = 0; v < 8; ++v) { c0[v] = bv0; c1[v] = bv1; }

        const float* Arow = A + (size_t)(row0 + n) * KDESC;
        for (int k = 0; k < KDESC; k += 4) {
            if ((k & 63) == 0) __builtin_prefetch(Arow + k + 128, 0, 3);  // global_prefetch_b8
            const int kr = k + hi * 2;
            v2f a = *(const v2f*)(Arow + kr);               // A: 16x4 f32 fragment
            v2f b0, b1;                                     // B: 4x16 f32 fragments
            b0[0] = W[kr * QQ + n];          b0[1] = W[(kr + 1) * QQ + n];
            b1[0] = W[kr * QQ + 16 + n];     b1[1] = W[(kr + 1) * QQ + 16 + n];
            c0 = wmma4(a, b0, c0);
            c1 = wmma4(a, b1, c1);
        }
#pragma unroll
        for (int v = 0; v < 8; ++v) {
            const int row = row0 + v + hi * 8;
            x[(size_t)row * DD + colBase + n]      = leakyf(c0[v]);
            x[(size_t)row * DD + colBase + 16 + n] = leakyf(c1[v]);
        }
    }

    // num/cat tails: 32 rows x 64 cols, K = 5 / 3 (too small for WMMA)
    for (int i = 0; i < 16; ++i) {
        const int idx = tid + i * 128;
        const int r   = idx >> 6;
        const int cc  = idx & 63;
        const int grow = rowBlk + r;
        if (grow >= NN) continue;
        float acc; int col;
        if (cc < 32) {
            acc = bn[cc];
#pragma unroll
            for (int k = 0; k < 5; ++k) acc += numf[(size_t)grow * 5 + k] * Wn[k * QQ + cc];
            col = 64 + cc;
        } else {
            const int c2 = cc - 32;
            acc = bc[c2];
#pragma unroll
            for (int k = 0; k < 3; ++k) acc += catf[(size_t)grow * 3 + k] * Wc[k * QQ + c2];
            col = 96 + c2;
        }
        x[(size_t)grow * DD + col] = leakyf(acc);
    }
}

// ---------------------------------------------------------------------------
// 128x128 WMMA GEMM with async-DMA LDS staging:
//   out = act(A@W [+ bias] [+ out_prev])
// A tile (16x128) and W K-chunks (32x128) are DMA'd global->LDS via
// GLOBAL_LOAD_ASYNC_TO_LDS_B128 (ASYNCcnt), WMMAs feed from the DS pipe.
// Padded LDS strides (132 / 144 floats) make all fragment reads bank-conflict-free.
// MAPPED_A: A holds mapped-uint max-agg values; unmap and zero ±inf (empty slots).
// ---------------------------------------------------------------------------
template <bool MAPPED_A, bool ACCUM, bool BIAS, bool ACT>
__global__ __launch_bounds__(128)
void gemm128_kernel(const float* __restrict__ A, const float* __restrict__ W,
                    const float* __restrict__ bias, float* __restrict__ out)
{
    constexpr int SAS = 132;   // sA row stride (mod 64 == 4)
    constexpr int SWS = 144;   // sW row stride (mod 64 == 16)
    __shared__ float sA[16 * SAS];   // ~8.4 KB
    __shared__ float sW[32 * SWS];   // ~18.4 KB

    const int tid  = threadIdx.x;
    const int lane = tid & 31;
    const int wid  = tid >> 5;
    const int row0 = blockIdx.x * 16;
    const int n  = lane & 15;
    const int hi = lane >> 4;
    const int colBase = wid * 32;

    // stage A tile: 16 rows x 32 float4 = 512 x 16B, 4 per thread
#pragma unroll
    for (int i = 0; i < 4; ++i) {
        const int idx = tid + i * 128;
        const int r = idx >> 5, c4 = idx & 31;
        async_copy16(A + (size_t)(row0 + r) * DD + c4 * 4, sA + r * SAS + c4 * 4);
    }

    v8f c0, c1;   // accumulator init overlaps with DMA
    if (ACCUM) {
#pragma unroll
        for (int v = 0; v < 8; ++v) {
            const int row = row0 + v + hi * 8;
            c0[v] = out[(size_t)row * DD + colBase + n];
            c1[v] = out[(size_t)row * DD + colBase + 16 + n];
        }
    } else if (BIAS) {
        float bv0 = bias[colBase + n], bv1 = bias[colBase + 16 + n];
#pragma unroll
        for (int v = 0; v < 8; ++v) { c0[v] = bv0; c1[v] = bv1; }
    } else {
#pragma unroll
        for (int v = 0; v < 8; ++v) { c0[v] = 0.f; c1[v] = 0.f; }
    }

    for (int kc = 0; kc < 4; ++kc) {
        // stage W K-chunk: rows [kc*32, kc*32+32) -> 32 rows x 32 float4, 8 per thread
#pragma unroll
        for (int i = 0; i < 8; ++i) {
            const int idx = tid + i * 128;
            const int r = idx >> 5, c4 = idx & 31;
            async_copy16(W + (size_t)(kc * 32 + r) * DD + c4 * 4, sW + r * SWS + c4 * 4);
        }
        wait_async0();
        __syncthreads();

#pragma unroll
        for (int kk = 0; kk < 8; ++kk) {
            const int kg = kc * 32 + kk * 4 + hi * 2;  // column in sA
            const int kr = kk * 4 + hi * 2;            // row in sW
            v2f a;
            if (MAPPED_A) {
                const unsigned* Au = (const unsigned*)sA + n * SAS + kg;
                float f0 = unmapf(Au[0]);
                float f1 = unmapf(Au[1]);
                a[0] = (fabsf(f0) == __builtin_inff()) ? 0.f : f0;
                a[1] = (fabsf(f1) == __builtin_inff()) ? 0.f : f1;
            } else {
                a = *(const v2f*)(sA + n * SAS + kg);            // ds_load_b64
            }
            v2f b0, b1;
            b0[0] = sW[kr * SWS + colBase + n];       b0[1] = sW[(kr + 1) * SWS + colBase + n];
            b1[0] = sW[kr * SWS + colBase + 16 + n];  b1[1] = sW[(kr + 1) * SWS + colBase + 16 + n];
            c0 = wmma4(a, b0, c0);
            c1 = wmma4(a, b1, c1);
        }
        __syncthreads();   // all waves done reading sW before next chunk overwrites it
    }

#pragma unroll
    for (int v = 0; v < 8; ++v) {
        const int row = row0 + v + hi * 8;
        float o0 = c0[v], o1 = c1[v];
        if (ACT) { o0 = leakyf(o0); o1 = leakyf(o1); }
        out[(size_t)row * DD + colBase + n]      = o0;
        out[(size_t)row * DD + colBase + 16 + n] = o1;
    }
}

// ---------------------------------------------------------------------------
__global__ __launch_bounds__(256)
void init_agg_kernel(unsigned* __restrict__ agg)
{
    const int i = blockIdx.x * 256 + threadIdx.x;   // uint4 granularity; grid sized exactly
    uint4 v = make_uint4(AGG_INIT, AGG_INIT, AGG_INIT, AGG_INIT);
    ((uint4*)agg)[i] = v;
}

// One wave per edge; lane covers 4 channels. No-return u32 atomic max (order-independent,
// replay-deterministic). Full 512B row read is perfectly coalesced.
__global__ __launch_bounds__(256)
void scatter_max_kernel(const int* __restrict__ edge_index, const int* __restrict__ edge_type,
                        const float* __restrict__ x, unsigned* __restrict__ agg, int rel)
{
    const int e = blockIdx.x * 8 + (threadIdx.x >> 5);
    if (e >= EE) return;
    if (edge_type[e] != rel) return;
    const int s = edge_index[e];
    const int d = edge_index[EE + e];
    const int lane = threadIdx.x & 31;
    const float4 v = *(const float4*)(x + (size_t)s * DD + lane * 4);
    unsigned* dstp = agg + (size_t)d * DD + lane * 4;
    atomicMax(dstp + 0, mapf(v.x));
    atomicMax(dstp + 1, mapf(v.y));
    atomicMax(dstp + 2, mapf(v.z));
    atomicMax(dstp + 3, mapf(v.w));
}

// ---------------------------------------------------------------------------
// Graph-LayerNorm: deterministic two-stage reduction over all N*D elements.
// ---------------------------------------------------------------------------
__global__ __launch_bounds__(256)
void ln_reduce1(const float* __restrict__ x, float* __restrict__ partials)
{
    __shared__ float ssum[256], ssq[256];
    float s = 0.f, q = 0.f;
    const int total = NN * DD / 4;
    for (int i = blockIdx.x * 256 + threadIdx.x; i < total; i += RBLK * 256) {
        float4 v = ((const float4*)x)[i];
        s += v.x + v.y + v.z + v.w;
        q += v.x * v.x + v.y * v.y + v.z * v.z + v.w * v.w;
    }
    ssum[threadIdx.x] = s; ssq[threadIdx.x] = q;
    __syncthreads();
    for (int st = 128; st > 0; st >>= 1) {
        if ((int)threadIdx.x < st) {
            ssum[threadIdx.x] += ssum[threadIdx.x + st];
            ssq[threadIdx.x]  += ssq[threadIdx.x + st];
        }
        __syncthreads();
    }
    if (threadIdx.x == 0) {
        partials[blockIdx.x]        = ssum[0];
        partials[RBLK + blockIdx.x] = ssq[0];
    }
}

__global__ __launch_bounds__(512)
void ln_reduce2(const float* __restrict__ partials, float* __restrict__ stats)
{
    __shared__ float ssum[512], ssq[512];
    const int t = threadIdx.x;
    ssum[t] = partials[t]; ssq[t] = partials[RBLK + t];
    __syncthreads();
    for (int st = 256; st > 0; st >>= 1) {
        if (t < st) { ssum[t] += ssum[t + st]; ssq[t] += ssq[t + st]; }
        __syncthreads();
    }
    if (t == 0) {
        const float inv = 1.0f / (float)(NN * DD);
        const float mu  = ssum[0] * inv;
        const float var = ssq[0] * inv - mu * mu;
        stats[0] = mu;
        stats[1] = rsqrtf(var + EPSV);
    }
}

__global__ __launch_bounds__(256)
void ln_apply(const float* __restrict__ xin, const float* __restrict__ stats,
              const float* __restrict__ g, const float* __restrict__ b,
              float* __restrict__ xout)
{
    const float mu = stats[0], rstd = stats[1];
    const int i = blockIdx.x * 256 + threadIdx.x;   // grid sized exactly
    const int col = i & (DD - 1);
    xout[i] = (xin[i] - mu) * rstd * g[col] + b[col];
}

// out[n] = sigmoid(x[n,:] . W3 + b3)
__global__ __launch_bounds__(256)
void head_kernel(const float* __restrict__ x, const float* __restrict__ W3,
                 const float* __restrict__ b3, float* __restrict__ out)
{
    const int nId = blockIdx.x * 256 + threadIdx.x;
    if (nId >= NN) return;
    const float4* xr = (const float4*)(x + (size_t)nId * DD);
    float acc = b3[0];
#pragma unroll
    for (int k = 0; k < 32; ++k) {
        float4 v = xr[k];
        acc += v.x * W3[4 * k] + v.y * W3[4 * k + 1] + v.z * W3[4 * k + 2] + v.w * W3[4 * k + 3];
    }
    out[nId] = 1.0f / (1.0f + __expf(-acc));
}

// ---------------------------------------------------------------------------
extern "C" void kernel_launch(void* const* d_in, const int* in_sizes, int n_in,
                              void* d_out, int out_size, void* d_ws, size_t ws_size,
                              hipStream_t stream)
{
    const float* desc = (const float*)d_in[0];
    const float* tw   = (const float*)d_in[1];
    const float* numf = (const float*)d_in[2];
    const float* catf = (const float*)d_in[3];
    const int* edge_index = (const int*)d_in[4];
    const int* edge_type  = (const int*)d_in[5];
    const float* Wd = (const float*)d_in[6];   const float* bd = (const float*)d_in[7];
    const float* Wt = (const float*)d_in[8];   const float* bt = (const float*)d_in[9];
    const float* Wn = (const float*)d_in[10];  const float* bn = (const float*)d_in[11];
    const float* Wc = (const float*)d_in[12];  const float* bc = (const float*)d_in[13];
    const float* W1 = (const float*)d_in[14];  const float* b1 = (const float*)d_in[15];
    const float* rg1_w    = (const float*)d_in[16];
    const float* rg1_root = (const float*)d_in[17];
    const float* rg1_bias = (const float*)d_in[18];
    const float* ln1_g = (const float*)d_in[19];
    const float* ln1_b = (const float*)d_in[20];
    const float* rg2_w    = (const float*)d_in[21];
    const float* rg2_root = (const float*)d_in[22];
    const float* rg2_bias = (const float*)d_in[23];
    const float* ln2_g = (const float*)d_in[24];
    const float* ln2_b = (const float*)d_in[25];
    const float* W2 = (const float*)d_in[26];  const float* b2 = (const float*)d_in[27];
    const float* W3 = (const float*)d_in[28];  const float* b3 = (const float*)d_in[29];
    float* out = (float*)d_out;

    // workspace layout
    float*    xA       = (float*)d_ws;
    float*    xB       = xA + (size_t)NN * DD;
    unsigned* agg      = (unsigned*)(xB + (size_t)NN * DD);
    float*    partials = (float*)(agg + (size_t)NN * DD);
    float*    stats    = partials + 2 * RBLK;

    const dim3 projGrid((NN + 31) / 32), b128(128);
    const dim3 gemmGrid(NN / 16);
    const dim3 aggGrid(NN * DD / 4 / 256), b256(256);
    const dim3 scatGrid((EE + 7) / 8);
    const dim3 lnaGrid(NN * DD / 256);

    // 1) fused projection -> xA
    proj_kernel<<<projGrid, b128, 0, stream>>>(desc, tw, numf, catf,
                                               Wd, bd, Wt, bt, Wn, bn, Wc, bc, xA);
    // 2) xB = leaky(xA @ W1 + b1)
    gemm128_kernel<false, false, true, true><<<gemmGrid, b128, 0, stream>>>(xA, W1, b1, xB);

    // 3) RGCN layer 1: xA = xB@root + bias + sum_r sanit(max_agg_r(xB)) @ W_r
    gemm128_kernel<false, false, true, false><<<gemmGrid, b128, 0, stream>>>(xB, rg1_root, rg1_bias, xA);
    for (int r = 0; r < 2; ++r) {
        init_agg_kernel<<<aggGrid, b256, 0, stream>>>(agg);
        scatter_max_kernel<<<scatGrid, b256, 0, stream>>>(edge_index, edge_type, xB, agg, r);
        gemm128_kernel<true, true, false, false><<<gemmGrid, b128, 0, stream>>>(
            (const float*)agg, rg1_w + (size_t)r * DD * DD, nullptr, xA);
    }
    // 4) graph layernorm 1: xA -> xB
    ln_reduce1<<<dim3(RBLK), b256, 0, stream>>>(xA, partials);
    ln_reduce2<<<dim3(1), dim3(512), 0, stream>>>(partials, stats);
    ln_apply<<<lnaGrid, b256, 0, stream>>>(xA, stats, ln1_g, ln1_b, xB);

    // 5) RGCN layer 2: xB -> xA
    gemm128_kernel<false, false, true, false><<<gemmGrid, b128, 0, stream>>>(xB, rg2_root, rg2_bias, xA);
    for (int r = 0; r < 2; ++r) {
        init_agg_kernel<<<aggGrid, b256, 0, stream>>>(agg);
        scatter_max_kernel<<<scatGrid, b256, 0, stream>>>(edge_index, edge_type, xB, agg, r);
        gemm128_kernel<true, true, false, false><<<gemmGrid, b128, 0, stream>>>(
            (const float*)agg, rg2_w + (size_t)r * DD * DD, nullptr, xA);
    }
    // 6) graph layernorm 2: xA -> xB
    ln_reduce1<<<dim3(RBLK), b256, 0, stream>>>(xA, partials);
    ln_reduce2<<<dim3(1), dim3(512), 0, stream>>>(partials, stats);
    ln_apply<<<lnaGrid, b256, 0, stream>>>(xA, stats, ln2_g, ln2_b, xB);

    // 7) xA = leaky(xB @ W2 + b2)
    gemm128_kernel<false, false, true, true><<<gemmGrid, b128, 0, stream>>>(xB, W2, b2, xA);
    // 8) head
    head_kernel<<<dim3((NN + 255) / 256), b256, 0, stream>>>(xA, W3, b3, out);
}